// MoETextProjection_71665824301088
// MI455X (gfx1250) — compile-verified
//
#include <hip/hip_runtime.h>
#include <hip/hip_bf16.h>

typedef __attribute__((ext_vector_type(2))) float v2f;
typedef __attribute__((ext_vector_type(8))) float v8f;

#define TOKENS   32768
#define D_IN     768
#define D_OUT    256
#define NEXP     8
#define BLK_M    32      // tokens per block (2 WMMA M-tiles)
#define LDS_STRIDE 772   // 768 + 4 pad dwords -> 16 lane-rows hit distinct banks

// ---------------------------------------------------------------------------
// Kernel 1: gating. One wave (32 lanes) per token. Dense weight vector out:
// wdense[t][e] = softmax weight if e in top-2, else 0.
// ---------------------------------------------------------------------------
__global__ __launch_bounds__(256) void moe_gate_kernel(
    const float* __restrict__ x,       // (TOKENS, D_IN)
    const float* __restrict__ gate_W,  // (D_IN, NEXP)
    const float* __restrict__ gate_b,  // (NEXP)
    float* __restrict__ wdense)        // (TOKENS, NEXP)
{
    const int wave  = threadIdx.x >> 5;
    const int lane  = threadIdx.x & 31;
    const int token = blockIdx.x * 8 + wave;
    if (token >= TOKENS) return;

    const float* xt = x + (size_t)token * D_IN;

    float acc[NEXP];
#pragma unroll
    for (int e = 0; e < NEXP; ++e) acc[e] = 0.0f;

    for (int k = lane; k < D_IN; k += 32) {
        const float xv = xt[k];
        const float* gw = gate_W + (size_t)k * NEXP;
#pragma unroll
        for (int e = 0; e < NEXP; ++e)
            acc[e] = fmaf(xv, gw[e], acc[e]);
    }

    // wave32 reduction
#pragma unroll
    for (int off = 16; off > 0; off >>= 1) {
#pragma unroll
        for (int e = 0; e < NEXP; ++e)
            acc[e] += __shfl_down(acc[e], off, 32);
    }

    if (lane == 0) {
        float logits[NEXP];
        float mx = -3.4e38f;
#pragma unroll
        for (int e = 0; e < NEXP; ++e) {
            logits[e] = acc[e] + gate_b[e];
            mx = fmaxf(mx, logits[e]);
        }
        float p[NEXP];
        float se = 0.0f;
#pragma unroll
        for (int e = 0; e < NEXP; ++e) {
            p[e] = __expf(logits[e] - mx);
            se += p[e];
        }
        const float inv = 1.0f / se;

        // top-2 (ties -> lowest index, matching lax.top_k)
        int i0 = 0; float v0 = p[0];
#pragma unroll
        for (int e = 1; e < NEXP; ++e)
            if (p[e] > v0) { v0 = p[e]; i0 = e; }
        int i1 = (i0 == 0) ? 1 : 0; float v1 = p[i1];
#pragma unroll
        for (int e = 0; e < NEXP; ++e)
            if (e != i0 && p[e] > v1) { v1 = p[e]; i1 = e; }

        float w[NEXP];
#pragma unroll
        for (int e = 0; e < NEXP; ++e) w[e] = 0.0f;
        w[i0] = v0 * inv;
        w[i1] = v1 * inv;

        float* wd = wdense + (size_t)token * NEXP;
#pragma unroll
        for (int e = 0; e < NEXP; ++e) wd[e] = w[e];
    }
}

// ---------------------------------------------------------------------------
// Kernel 2: weighted all-expert projection via V_WMMA_F32_16X16X4_F32.
// Block: 256 threads = 8 waves; tile = 32 tokens x 256 outputs.
// Wave w owns N columns [32w, 32w+32) as two 16x16 N-tiles, and computes BOTH
// 16-token M-tiles so each B operand feeds two WMMAs (halves B traffic/FLOP).
// out[t][n] = sum_e w[t][e] * ( (x @ W_e)[t][n] + b_e[n] )
// ---------------------------------------------------------------------------
__global__ __launch_bounds__(256) void moe_expert_kernel(
    const float* __restrict__ x,         // (TOKENS, D_IN)
    const float* __restrict__ expert_W,  // (NEXP, D_IN, D_OUT) row-major [e][k][n]
    const float* __restrict__ expert_b,  // (NEXP, D_OUT)
    const float* __restrict__ wdense,    // (TOKENS, NEXP)
    float* __restrict__ out)             // (TOKENS, D_OUT)
{
    __shared__ float lA[BLK_M * LDS_STRIDE];   // 32-token A tile, padded rows
    __shared__ float lW[BLK_M * NEXP];         // routing weights for these tokens

    const int block_m = blockIdx.x * BLK_M;    // first token of this tile
    const int tid     = threadIdx.x;

    // Cooperative, coalesced A-tile load: 32 x 768 fp32 via float4.
    {
        const float4* src = reinterpret_cast<const float4*>(x + (size_t)block_m * D_IN);
#pragma unroll
        for (int i = 0; i < (BLK_M * D_IN / 4) / 256; ++i) {
            const int idx = i * 256 + tid;
            const int row = idx / (D_IN / 4);
            const int c4  = idx % (D_IN / 4);
            const float4 v = src[(size_t)row * (D_IN / 4) + c4];
            float* dst = &lA[row * LDS_STRIDE + c4 * 4];
            dst[0] = v.x; dst[1] = v.y; dst[2] = v.z; dst[3] = v.w;
        }
        lW[tid] = wdense[(size_t)block_m * NEXP + tid];   // 256 contiguous floats
    }
    __syncthreads();

    const int lane   = tid & 31;
    const int laneM  = lane & 15;   // row within M-tile / column within N-tile
    const int laneHi = lane >> 4;   // K-half selector
    const int wave   = tid >> 5;
    const int n0     = wave * 32;   // this wave's first output column

    v8f acc00 = {};  // M-tile 0, N-tile 0
    v8f acc01 = {};  // M-tile 0, N-tile 1
    v8f acc10 = {};  // M-tile 1, N-tile 0
    v8f acc11 = {};  // M-tile 1, N-tile 1

    for (int e = 0; e < NEXP; ++e) {
        // Init WMMA accumulators with the expert bias (depends only on N=lane).
        const float bias0 = expert_b[e * D_OUT + n0 + laneM];
        const float bias1 = expert_b[e * D_OUT + n0 + 16 + laneM];
        v8f c00, c01, c10, c11;
#pragma unroll
        for (int r = 0; r < 8; ++r) {
            c00[r] = bias0; c01[r] = bias1;
            c10[r] = bias0; c11[r] = bias1;
        }

        const float* We = expert_W + (size_t)e * D_IN * D_OUT;

#pragma unroll 2
        for (int k0 = 0; k0 < D_IN; k0 += 4) {
            // A 16x4 fp32 per M-tile: v0 = A[M][k0+2*laneHi], v1 = A[M][k0+2*laneHi+1]
            const float* ap0 = &lA[laneM * LDS_STRIDE + k0 + 2 * laneHi];
            const float* ap1 = ap0 + 16 * LDS_STRIDE;
            v2f a0; a0.x = ap0[0]; a0.y = ap0[1];      // ds_load_b64
            v2f a1; a1.x = ap1[0]; a1.y = ap1[1];      // ds_load_b64

            // B 4x16 fp32: v0 = B[k0+2*laneHi][n], v1 = B[k0+2*laneHi+1][n]
            const float* bp0 = We + (size_t)(k0 + 2 * laneHi) * D_OUT + n0 + laneM;
            v2f b0; b0.x = bp0[0];  b0.y = bp0[D_OUT];
            const float* bp1 = bp0 + 16;
            v2f b1; b1.x = bp1[0];  b1.y = bp1[D_OUT];

            // Each B operand feeds two WMMAs (both M-tiles).
            c00 = __builtin_amdgcn_wmma_f32_16x16x4_f32(
                      false, a0, false, b0, (short)0, c00, false, false);
            c10 = __builtin_amdgcn_wmma_f32_16x16x4_f32(
                      false, a1, false, b0, (short)0, c10, false, false);
            c01 = __builtin_amdgcn_wmma_f32_16x16x4_f32(
                      false, a0, false, b1, (short)0, c01, false, false);
            c11 = __builtin_amdgcn_wmma_f32_16x16x4_f32(
                      false, a1, false, b1, (short)0, c11, false, false);
        }

        // Fold into final accumulators, scaled by per-row routing weight.
        // C layout: VGPR r -> M = r + 8*laneHi (+16 for M-tile 1).
#pragma unroll
        for (int r = 0; r < 8; ++r) {
            const float w0 = lW[(r + 8 * laneHi) * NEXP + e];
            const float w1 = lW[(16 + r + 8 * laneHi) * NEXP + e];
            acc00[r] = fmaf(w0, c00[r], acc00[r]);
            acc01[r] = fmaf(w0, c01[r], acc01[r]);
            acc10[r] = fmaf(w1, c10[r], acc10[r]);
            acc11[r] = fmaf(w1, c11[r], acc11[r]);
        }
    }

    // Store: row M = mt*16 + r + 8*laneHi, col N = n0 + laneM (+16 for N-tile 1).
#pragma unroll
    for (int r = 0; r < 8; ++r) {
        const int m0 = block_m + r + 8 * laneHi;
        const int m1 = m0 + 16;
        out[(size_t)m0 * D_OUT + n0 + laneM]      = acc00[r];
        out[(size_t)m0 * D_OUT + n0 + 16 + laneM] = acc01[r];
        out[(size_t)m1 * D_OUT + n0 + laneM]      = acc10[r];
        out[(size_t)m1 * D_OUT + n0 + 16 + laneM] = acc11[r];
    }
}

// ---------------------------------------------------------------------------
extern "C" void kernel_launch(void* const* d_in, const int* in_sizes, int n_in,
                              void* d_out, int out_size, void* d_ws, size_t ws_size,
                              hipStream_t stream) {
    const float* x        = (const float*)d_in[0];  // (32,1024,768)
    const float* gate_W   = (const float*)d_in[1];  // (768,8)
    const float* gate_b   = (const float*)d_in[2];  // (8)
    const float* expert_W = (const float*)d_in[3];  // (8,768,256)
    const float* expert_b = (const float*)d_in[4];  // (8,256)
    float* out = (float*)d_out;                     // (32,1024,256)

    float* wdense = (float*)d_ws;                   // (TOKENS, 8) = 1 MB

    // Gate: one wave per token, 8 tokens per 256-thread block.
    moe_gate_kernel<<<TOKENS / 8, 256, 0, stream>>>(x, gate_W, gate_b, wdense);

    // Expert GEMM + weighted combine: 32 tokens per block.
    moe_expert_kernel<<<TOKENS / BLK_M, 256, 0, stream>>>(x, expert_W, expert_b,
                                                          wdense, out);
}